// Gen_QueryAndGroupXYZ_88278757802044
// MI455X (gfx1250) — compile-verified
//
#include <hip/hip_runtime.h>

typedef __attribute__((ext_vector_type(2))) float v2f;
typedef __attribute__((ext_vector_type(8))) float v8f;

#define BATCH 4
#define NPTS  16384
#define NQ    4096
#define KNN   32
#define CHUNK 2048      // points per staged chunk (24 KB raw xyz)
#define NCHUNK (NPTS / CHUNK)
#define WAVES 8         // waves per block
#define QPW   16        // queries per wave (one WMMA row-tile)
#define CPYS  6         // async b128 copies per thread per chunk (6*256*16 = 24576 B)

// Insert candidate (cd, ci) into ascending sorted register list via unrolled
// compare-swap chain. Strict '<' keeps earliest (lowest index) on ties,
// matching top_k's stable ordering.
__device__ __forceinline__ void insert32(float (&d)[KNN], int (&ix)[KNN],
                                         float cd, int ci) {
  if (cd < d[KNN - 1]) {
#pragma unroll
    for (int s = 0; s < KNN; ++s) {
      bool sw = cd < d[s];
      float td = d[s]; int ti = ix[s];
      d[s]  = sw ? cd : td;
      ix[s] = sw ? ci : ti;
      cd    = sw ? td : cd;
      ci    = sw ? ti : ci;
    }
  }
}

__global__ __launch_bounds__(256)
void knn_group_kernel(const float* __restrict__ xyz,
                      const float* __restrict__ new_xyz,
                      float* __restrict__ out) {
  __shared__ float s_raw[2][CHUNK * 3];      // double-buffered raw xyz (2 x 24 KB)
  __shared__ float s_keys[WAVES][16][16];    // per-wave WMMA tile transpose (8 KB)

  const int t    = threadIdx.x;
  const int wave = t >> 5;
  const int lane = t & 31;
  const int half = lane >> 4;   // 0: lanes 0-15, 1: lanes 16-31
  const int r    = lane & 15;

  const int b      = blockIdx.x >> 5;              // 32 blocks per batch
  const int qblock = (blockIdx.x & 31) * (WAVES * QPW);
  const int m      = qblock + wave * QPW + r;      // query served by this lane

  // ---- A fragment (constant for the whole kernel) ----
  // Row M of A = (-2qx, -2qy, -2qz, 1); key = A.B = |p|^2 - 2 q.p
  // (row-constant |q|^2 dropped: it cannot change per-query ordering)
  // 16x4 f32 A layout: lane l holds M=l&15; vgpr0 = K(half?2:0), vgpr1 = K(half?3:1)
  const float* q = new_xyz + ((size_t)b * NQ + m) * 3;
  float qx = q[0], qy = q[1], qz = q[2];
  v2f a;
  a[0] = half ? (-2.0f * qz) : (-2.0f * qx);
  a[1] = half ? 1.0f         : (-2.0f * qy);

  float dlist[KNN]; int ilist[KNN];
#pragma unroll
  for (int i = 0; i < KNN; ++i) { dlist[i] = 3.4e38f; ilist[i] = 0; }

  const float* xb = xyz + (size_t)b * NPTS * 3;

  // ---- async stage of one chunk: 24576 bytes via 6 x b128 per thread ----
  auto async_stage = [&](int chunk_idx) {
    unsigned long long gbase =
        (unsigned long long)(const void*)(xb + (size_t)chunk_idx * CHUNK * 3);
    unsigned int lbase =
        (unsigned int)(unsigned long long)(const void*)&s_raw[chunk_idx & 1][0];
#pragma unroll
    for (int i = 0; i < CPYS; ++i) {
      unsigned int off  = (unsigned int)(i * 256 + t) * 16u;
      unsigned int laddr = lbase + off;
      asm volatile("global_load_async_to_lds_b128 %0, %1, %2"
                   :: "v"(laddr), "v"(off), "s"(gbase)
                   : "memory");
    }
  };

  async_stage(0);   // prologue: kick off chunk 0

  for (int c = 0; c < NCHUNK; ++c) {
    // Prefetch next chunk into the other buffer (safe: that buffer's last
    // readers synced at the previous iteration's trailing barrier), then
    // drain to <= CPYS outstanding so the *current* chunk is complete.
    if (c + 1 < NCHUNK) {
      async_stage(c + 1);
      asm volatile("s_wait_asynccnt 0x6" ::: "memory");
    } else {
      asm volatile("s_wait_asynccnt 0x0" ::: "memory");
    }
    __syncthreads();

    const float* buf = s_raw[c & 1];
    const int cb = c * CHUNK;

    for (int g = 0; g < CHUNK / 16; ++g) {
      // ---- B fragment: 4x16 (px,py,pz,p2) for 16 points from LDS ----
      // lane l holds N=l&15; vgpr0 = K(half?2:0), vgpr1 = K(half?3:1)
      int lp = g * 16 + r;
      float f0 = buf[lp * 3 + 0];
      float f1 = buf[lp * 3 + 1];
      float f2 = buf[lp * 3 + 2];
      float p2v = fmaf(f0, f0, fmaf(f1, f1, f2 * f2));
      v2f bf;
      bf[0] = half ? f2  : f0;
      bf[1] = half ? p2v : f1;

      v8f cacc = {};
      cacc = __builtin_amdgcn_wmma_f32_16x16x4_f32(
                 false, a, false, bf, (short)0, cacc, false, false);

      // ---- transpose 16x16 key tile through per-wave LDS scratch ----
      // C layout: lane l, vgpr v -> M = v + 8*half, N = l&15
#pragma unroll
      for (int v = 0; v < 8; ++v)
        s_keys[wave][half * 8 + v][r] = cacc[v];
      asm volatile("s_wait_dscnt 0x0" ::: "memory");
      const float4* kp = (const float4*)&s_keys[wave][r][half * 8];
      float4 k0 = kp[0];
      float4 k1 = kp[1];

      // Group-level early-out: one EXEC branch instead of eight.
      float kmin = fminf(fminf(fminf(k0.x, k0.y), fminf(k0.z, k0.w)),
                         fminf(fminf(k1.x, k1.y), fminf(k1.z, k1.w)));
      if (kmin < dlist[KNN - 1]) {
        int base = cb + g * 16 + half * 8;
        insert32(dlist, ilist, k0.x, base + 0);
        insert32(dlist, ilist, k0.y, base + 1);
        insert32(dlist, ilist, k0.z, base + 2);
        insert32(dlist, ilist, k0.w, base + 3);
        insert32(dlist, ilist, k1.x, base + 4);
        insert32(dlist, ilist, k1.y, base + 5);
        insert32(dlist, ilist, k1.z, base + 6);
        insert32(dlist, ilist, k1.w, base + 7);
      }
    }
    __syncthreads();
  }

  // ---- merge partner halves (lane ^ 16): bitonic lower-half + cleanup ----
  float md[KNN]; int mi[KNN];
#pragma unroll
  for (int i = 0; i < KNN; ++i) {
    float od = __shfl_xor(dlist[KNN - 1 - i], 16, 32);
    int   oi = __shfl_xor(ilist[KNN - 1 - i], 16, 32);
    bool take = dlist[i] <= od;
    md[i] = take ? dlist[i] : od;
    mi[i] = take ? ilist[i] : oi;
  }
#pragma unroll
  for (int s = 16; s >= 1; s >>= 1) {
#pragma unroll
    for (int i = 0; i < KNN; ++i) {
      if ((i & s) == 0) {
        int j = i + s;
        bool sw = md[j] < md[i];
        float td = md[i]; int ti = mi[i];
        md[i] = sw ? md[j] : td;  mi[i] = sw ? mi[j] : ti;
        md[j] = sw ? td : md[j];  mi[j] = sw ? ti : mi[j];
      }
    }
  }

  // ---- gather grouped coordinates: out[b][c][m][k] = xyz[b][mi[k]][c] ----
  if (half == 0) {
    float* ox = out + (((size_t)b * 3 + 0) * NQ + m) * KNN;
    float* oy = out + (((size_t)b * 3 + 1) * NQ + m) * KNN;
    float* oz = out + (((size_t)b * 3 + 2) * NQ + m) * KNN;
#pragma unroll
    for (int k = 0; k < KNN; k += 4) {
      const float* p0 = xb + (size_t)mi[k + 0] * 3;
      const float* p1 = xb + (size_t)mi[k + 1] * 3;
      const float* p2 = xb + (size_t)mi[k + 2] * 3;
      const float* p3 = xb + (size_t)mi[k + 3] * 3;
      float4 vx = make_float4(p0[0], p1[0], p2[0], p3[0]);
      float4 vy = make_float4(p0[1], p1[1], p2[1], p3[1]);
      float4 vz = make_float4(p0[2], p1[2], p2[2], p3[2]);
      *(float4*)(ox + k) = vx;
      *(float4*)(oy + k) = vy;
      *(float4*)(oz + k) = vz;
    }
  }
}

extern "C" void kernel_launch(void* const* d_in, const int* in_sizes, int n_in,
                              void* d_out, int out_size, void* d_ws, size_t ws_size,
                              hipStream_t stream) {
  const float* xyz     = (const float*)d_in[0];
  const float* new_xyz = (const float*)d_in[1];
  float* out = (float*)d_out;
  dim3 grid(BATCH * (NQ / (WAVES * QPW)));   // 4 * 32 = 128 blocks
  knn_group_kernel<<<grid, 256, 0, stream>>>(xyz, new_xyz, out);
}